// DataTaskGAT2Layer_60318520705357
// MI455X (gfx1250) — compile-verified
//
#include <hip/hip_runtime.h>
#include <math.h>

typedef float v2f __attribute__((ext_vector_type(2)));
typedef float v8f __attribute__((ext_vector_type(8)));

#define LRELU(x, s) ((x) > 0.f ? (x) : (s) * (x))

// ---------------------------------------------------------------------------
// Utility fills
// ---------------------------------------------------------------------------
__global__ __launch_bounds__(256) void fill_kernel(float* p, float v, long long n) {
  long long i = (long long)blockIdx.x * blockDim.x + threadIdx.x;
  if (i < n) p[i] = v;
}

// ---------------------------------------------------------------------------
// V[k,h] = sum_c W[k, h*16+c] * att[h,c]   (collapses att-dot into the GEMM)
// ---------------------------------------------------------------------------
__global__ __launch_bounds__(256) void compute_V_kernel(const float* __restrict__ Wm,
                                                        const float* __restrict__ att,
                                                        float* __restrict__ V, int K) {
  int t = blockIdx.x * blockDim.x + threadIdx.x;
  if (t >= K * 4) return;
  int k = t >> 2, h = t & 3;
  float s = 0.f;
  for (int c = 0; c < 16; ++c) s += Wm[k * 64 + h * 16 + c] * att[h * 16 + c];
  V[t] = s;
}

// out[n,h] = sum_k X[n,k] * V[k,h]
__global__ __launch_bounds__(256) void node_att_kernel(const float* __restrict__ X,
                                                       const float* __restrict__ V,
                                                       float* __restrict__ out, int Nn, int K) {
  int n = blockIdx.x * blockDim.x + threadIdx.x;
  if (n >= Nn) return;
  float a0 = 0.f, a1 = 0.f, a2 = 0.f, a3 = 0.f;
  for (int k = 0; k < K; ++k) {
    float xv = X[(size_t)n * K + k];
    a0 += xv * V[k * 4 + 0];
    a1 += xv * V[k * 4 + 1];
    a2 += xv * V[k * 4 + 2];
    a3 += xv * V[k * 4 + 3];
  }
  out[(size_t)n * 4 + 0] = a0;
  out[(size_t)n * 4 + 1] = a1;
  out[(size_t)n * 4 + 2] = a2;
  out[(size_t)n * 4 + 3] = a3;
}

// ---------------------------------------------------------------------------
// f32 GEMM via V_WMMA_F32_16X16X4_F32:  Out[M,N] = X[M,K] @ W[K,N]
//   K, N compile-time: fully unrolled, no branches around WMMA.
//   One wave per 16-row stripe across all N; 8 waves per block; W in LDS.
//   Full-tile fast path: unguarded, coalesced stores (M here is always a
//   multiple of 16, so the guarded remainder path is cold).
// A layout (16x4 f32): lane&15 = M; lanes 0-15 -> K={0,1}, lanes 16-31 -> K={2,3}
// B layout (4x16 f32): lane&15 = N; VGPR v holds K = v + 2*(lane>=16)
// C/D layout (16x16 f32): VGPR r: lanes 0-15 -> M=r, lanes 16-31 -> M=r+8
// ---------------------------------------------------------------------------
template <int K, int N>
__global__ __launch_bounds__(256) void gemm_wmma_kernel(const float* __restrict__ X,
                                                        const float* __restrict__ W,
                                                        float* __restrict__ Out, int M) {
  constexpr int K4 = (K + 3) / 4;   // k-steps
  constexpr int KP = K4 * 4;        // padded K
  constexpr int NT = N / 16;        // n-tiles

  __shared__ float sW[KP * N];
  int tid = threadIdx.x;
  for (int i = tid; i < KP * N; i += 256) {
    int k = i / N;
    sW[i] = (k < K) ? W[i] : 0.f;
  }
  __syncthreads();

  int lane = tid & 31;
  int laneM = lane & 15;  // row within A tile / column within B,C,D tiles
  int half = lane >> 4;
  long long m0 = ((long long)blockIdx.x * 8 + (tid >> 5)) * 16;
  if (m0 >= M) return;  // uniform per wave -> EXEC all-ones at WMMA

  bool fullTile = (m0 + 16 <= M);  // wave-uniform
  long long row = m0 + laneM;
  bool rowOK = row < M;

  // A fragments for the whole 16-row stripe (reused across all n-tiles)
  v2f a[K4];
#pragma unroll
  for (int kk = 0; kk < K4; ++kk) {
    int kb = kk * 4 + half * 2;
    a[kk].x = (rowOK && kb < K) ? X[row * K + kb] : 0.f;
    a[kk].y = (rowOK && (kb + 1) < K) ? X[row * K + kb + 1] : 0.f;
  }

  v8f acc[NT];
#pragma unroll
  for (int nt = 0; nt < NT; ++nt) acc[nt] = (v8f){0.f, 0.f, 0.f, 0.f, 0.f, 0.f, 0.f, 0.f};

#pragma unroll
  for (int kk = 0; kk < K4; ++kk) {
    int kb = kk * 4 + half * 2;
#pragma unroll
    for (int nt = 0; nt < NT; ++nt) {
      int ncol = nt * 16 + laneM;
      v2f b;
      b.x = sW[kb * N + ncol];
      b.y = sW[(kb + 1) * N + ncol];
      acc[nt] = __builtin_amdgcn_wmma_f32_16x16x4_f32(false, a[kk], false, b, (short)0,
                                                      acc[nt], false, false);
    }
  }

  // Store epilogue. Fast path: no per-element bounds checks / exec juggling.
  float* base = Out + (m0 + half * 8) * (long long)N + laneM;  // row m0+half*8, col laneM
  if (fullTile) {
#pragma unroll
    for (int nt = 0; nt < NT; ++nt) {
      float* p = base + nt * 16;
#pragma unroll
      for (int r = 0; r < 8; ++r) p[r * (long long)N] = acc[nt][r];
    }
  } else {
#pragma unroll
    for (int nt = 0; nt < NT; ++nt) {
      float* p = base + nt * 16;
#pragma unroll
      for (int r = 0; r < 8; ++r) {
        if (m0 + half * 8 + r < M) p[r * (long long)N] = acc[nt][r];
      }
    }
  }
}

static inline int cdiv(long long a, long long b) { return (int)((a + b - 1) / b); }

static void launch_gemm(const float* X, const float* W, float* Out, int M, int K, int N,
                        hipStream_t stream) {
  int grid = cdiv(M, 128);
  if (N == 64) {
    if (K == 5)
      gemm_wmma_kernel<5, 64><<<grid, 256, 0, stream>>>(X, W, Out, M);
    else if (K == 12)
      gemm_wmma_kernel<12, 64><<<grid, 256, 0, stream>>>(X, W, Out, M);
    else
      gemm_wmma_kernel<64, 64><<<grid, 256, 0, stream>>>(X, W, Out, M);
  } else {
    gemm_wmma_kernel<64, 16><<<grid, 256, 0, stream>>>(X, W, Out, M);
  }
}

// ---------------------------------------------------------------------------
// Edge passes (segment softmax + scatter)
// ---------------------------------------------------------------------------
__device__ inline void atomicMaxF(float* addr, float v) {
  // sign-split trick; buffer initialized to -inf
  if (v >= 0.f)
    atomicMax((int*)addr, __float_as_int(v));
  else
    atomicMin((unsigned int*)addr, __float_as_uint(v));
}

__global__ __launch_bounds__(256) void edge_alpha_max_kernel(
    const int* __restrict__ src, const int* __restrict__ dst, const float* __restrict__ ea,
    const float* __restrict__ Ve, const float* __restrict__ aS, const float* __restrict__ aD,
    float* __restrict__ alpha, float* __restrict__ amax, int E) {
  int e = blockIdx.x * blockDim.x + threadIdx.x;
  if (e >= E) return;
  int s = src[e], d = dst[e];
  float e0 = ea[(size_t)e * 3 + 0];
  float e1 = ea[(size_t)e * 3 + 1];
  float e2 = ea[(size_t)e * 3 + 2];
#pragma unroll
  for (int h = 0; h < 4; ++h) {
    float ae = e0 * Ve[0 * 4 + h] + e1 * Ve[1 * 4 + h] + e2 * Ve[2 * 4 + h];
    float al = aS[(size_t)s * 4 + h] + aD[(size_t)d * 4 + h] + ae;
    al = LRELU(al, 0.2f);
    alpha[(size_t)e * 4 + h] = al;
    atomicMaxF(&amax[(size_t)d * 4 + h], al);
  }
}

__global__ __launch_bounds__(256) void edge_exp_sum_kernel(const int* __restrict__ dst,
                                                           float* __restrict__ al_ex,
                                                           const float* __restrict__ amax,
                                                           float* __restrict__ denom, int E) {
  int e = blockIdx.x * blockDim.x + threadIdx.x;
  if (e >= E) return;
  int d = dst[e];
#pragma unroll
  for (int h = 0; h < 4; ++h) {
    float x = expf(al_ex[(size_t)e * 4 + h] - amax[(size_t)d * 4 + h]);
    al_ex[(size_t)e * 4 + h] = x;
    atomicAdd(&denom[(size_t)d * 4 + h], x);
  }
}

// 64 threads per edge: f = h*16+c; coalesced hs[src] loads + acc[dst] atomics
__global__ __launch_bounds__(256) void edge_scatter_kernel(
    const int* __restrict__ src, const int* __restrict__ dst, const float* __restrict__ hs,
    const float* __restrict__ ex, const float* __restrict__ denom, float* __restrict__ acc,
    long long E) {
  long long t = (long long)blockIdx.x * blockDim.x + threadIdx.x;
  long long e = t >> 6;
  int f = (int)(t & 63);
  if (e >= E) return;
  int s = src[e], d = dst[e];
  int h = f >> 4;
  float a = ex[e * 4 + h] / (denom[(size_t)d * 4 + h] + 1e-16f);
  atomicAdd(&acc[(size_t)d * 64 + f], hs[(size_t)s * 64 + f] * a);
}

// ---------------------------------------------------------------------------
// Epilogues
// ---------------------------------------------------------------------------
// concat layer: out = leaky(LN(acc + bias), 0.01); wave (32 lanes) per node, 2 feats/lane
__global__ __launch_bounds__(256) void finalize_ln_kernel(
    const float* __restrict__ acc, const float* __restrict__ bias, const float* __restrict__ g,
    const float* __restrict__ bln, float* __restrict__ out, int Nd) {
  long long t = (long long)blockIdx.x * blockDim.x + threadIdx.x;
  long long node = t >> 5;
  int lane = (int)(t & 31);
  if (node >= Nd) return;
  float v0 = acc[node * 64 + lane] + bias[lane];
  float v1 = acc[node * 64 + lane + 32] + bias[lane + 32];
  float s = v0 + v1, sq = v0 * v0 + v1 * v1;
  for (int off = 16; off > 0; off >>= 1) {
    s += __shfl_xor(s, off, 32);
    sq += __shfl_xor(sq, off, 32);
  }
  float mu = s * (1.f / 64.f);
  float var = sq * (1.f / 64.f) - mu * mu;
  float inv = rsqrtf(var + 1e-5f);
  float y0 = (v0 - mu) * inv * g[lane] + bln[lane];
  float y1 = (v1 - mu) * inv * g[lane + 32] + bln[lane + 32];
  out[node * 64 + lane] = LRELU(y0, 0.01f);
  out[node * 64 + lane + 32] = LRELU(y1, 0.01f);
}

// mean layer: out[n,c] = leaky(mean_h acc[n,h,c] + res[n,c] + bias[c], 0.01)
__global__ __launch_bounds__(256) void finalize_mean_kernel(const float* __restrict__ acc,
                                                            const float* __restrict__ res,
                                                            const float* __restrict__ bias,
                                                            float* __restrict__ out, int Nd) {
  long long t = (long long)blockIdx.x * blockDim.x + threadIdx.x;
  long long n = t >> 4;
  int c = (int)(t & 15);
  if (n >= Nd) return;
  float m = 0.f;
#pragma unroll
  for (int h = 0; h < 4; ++h) m += acc[n * 64 + h * 16 + c];
  m *= 0.25f;
  float y = m + res[n * 16 + c] + bias[c];
  out[n * 16 + c] = LRELU(y, 0.01f);
}

// ---------------------------------------------------------------------------
// Host-side driver
// ---------------------------------------------------------------------------
struct ConvWS {
  float *hs, *aS, *aD, *alpha, *amax, *denom, *acc, *res16, *Vsrc, *Vdst, *Vedge;
};

static void run_conv(const float* x_src, int Ns, int Ks, const float* x_dst, int Nd, int Kd,
                     const int* src, const int* dst, const float* ea, int E,
                     const float* W_src, const float* W_dst, const float* att_src,
                     const float* att_dst, const float* W_edge, const float* att_edge,
                     const float* W_res, const float* bias, bool concat, const float* ln_g,
                     const float* ln_b, float* out, const ConvWS& B, hipStream_t stream) {
  // collapsed attention vectors
  compute_V_kernel<<<1, 256, 0, stream>>>(W_src, att_src, B.Vsrc, Ks);
  compute_V_kernel<<<1, 256, 0, stream>>>(W_dst, att_dst, B.Vdst, Kd);
  compute_V_kernel<<<1, 64, 0, stream>>>(W_edge, att_edge, B.Vedge, 3);
  // hs = x_src @ W_src (WMMA)
  launch_gemm(x_src, W_src, B.hs, Ns, Ks, 64, stream);
  // attention logits per node
  node_att_kernel<<<cdiv(Ns, 256), 256, 0, stream>>>(x_src, B.Vsrc, B.aS, Ns, Ks);
  node_att_kernel<<<cdiv(Nd, 256), 256, 0, stream>>>(x_dst, B.Vdst, B.aD, Nd, Kd);
  // residual: seed accumulator (concat) or side buffer (mean)
  if (concat) {
    launch_gemm(x_dst, W_res, B.acc, Nd, Kd, 64, stream);
  } else {
    launch_gemm(x_dst, W_res, B.res16, Nd, Kd, 16, stream);
    fill_kernel<<<cdiv((long long)Nd * 64, 256), 256, 0, stream>>>(B.acc, 0.f, (long long)Nd * 64);
  }
  fill_kernel<<<cdiv((long long)Nd * 4, 256), 256, 0, stream>>>(B.amax, -INFINITY,
                                                                (long long)Nd * 4);
  fill_kernel<<<cdiv((long long)Nd * 4, 256), 256, 0, stream>>>(B.denom, 0.f, (long long)Nd * 4);
  // segment softmax
  edge_alpha_max_kernel<<<cdiv(E, 256), 256, 0, stream>>>(src, dst, ea, B.Vedge, B.aS, B.aD,
                                                          B.alpha, B.amax, E);
  edge_exp_sum_kernel<<<cdiv(E, 256), 256, 0, stream>>>(dst, B.alpha, B.amax, B.denom, E);
  // weighted message scatter
  edge_scatter_kernel<<<cdiv((long long)E * 64, 256), 256, 0, stream>>>(src, dst, B.hs, B.alpha,
                                                                        B.denom, B.acc, E);
  // epilogue
  if (concat)
    finalize_ln_kernel<<<cdiv((long long)Nd * 32, 256), 256, 0, stream>>>(B.acc, bias, ln_g, ln_b,
                                                                          out, Nd);
  else
    finalize_mean_kernel<<<cdiv((long long)Nd * 16, 256), 256, 0, stream>>>(B.acc, B.res16, bias,
                                                                            out, Nd);
}

extern "C" void kernel_launch(void* const* d_in, const int* in_sizes, int n_in, void* d_out,
                              int out_size, void* d_ws, size_t ws_size, hipStream_t stream) {
  const int TASK_F = 12, DATA_F = 5;
  const float* x_tasks = (const float*)d_in[0];
  const float* x_data = (const float*)d_in[1];
  const int* ei_dt = (const int*)d_in[2];  // [2,E]: row0 = src(data), row1 = dst(tasks)
  const int* ei_td = (const int*)d_in[3];  // [2,E]: row0 = src(tasks), row1 = dst(data)
  const float* ea_dt = (const float*)d_in[4];
  const float* ea_td = (const float*)d_in[5];
  const int NT = in_sizes[0] / TASK_F;
  const int ND = in_sizes[1] / DATA_F;
  const int E = in_sizes[2] / 2;

  // params flattened in dict order: dt1, td1, td2, dt2 (8 tensors each), ln_tasks(g,b), ln_data(g,b)
  const float* P[32];
  for (int i = 0; i < 32; ++i) P[i] = (const float*)d_in[6 + i];
  const float* ln_tg = (const float*)d_in[38];
  const float* ln_tb = (const float*)d_in[39];
  const float* ln_dg = (const float*)d_in[40];
  const float* ln_db = (const float*)d_in[41];

  // workspace carve-up (floats)
  float* w = (float*)d_ws;
  long long off = 0;
  ConvWS B;
  B.hs = w + off;    off += (long long)NT * 64;   // max(Ns)*64
  B.aS = w + off;    off += (long long)NT * 4;
  B.aD = w + off;    off += (long long)NT * 4;
  B.alpha = w + off; off += (long long)E * 4;
  B.amax = w + off;  off += (long long)NT * 4;
  B.denom = w + off; off += (long long)NT * 4;
  B.acc = w + off;   off += (long long)NT * 64;   // max(Nd)*64
  B.res16 = w + off; off += (long long)NT * 16;
  B.Vsrc = w + off;  off += 256;
  B.Vdst = w + off;  off += 256;
  B.Vedge = w + off; off += 64;
  float* t1 = w + off; off += (long long)NT * 64;
  float* d1 = w + off; off += (long long)ND * 64;

  float* t2 = (float*)d_out;                       // [NT,16] first in return order
  float* d2 = (float*)d_out + (long long)NT * 16;  // [ND,16]

  // layer 1 (concat=True, LN + leaky 0.01)
  // dt1: src=data, dst=tasks
  run_conv(x_data, ND, DATA_F, x_tasks, NT, TASK_F, ei_dt, ei_dt + E, ea_dt, E,
           P[0], P[1], P[2], P[3], P[4], P[5], P[6], P[7], true, ln_tg, ln_tb, t1, B, stream);
  // td1: src=tasks, dst=data
  run_conv(x_tasks, NT, TASK_F, x_data, ND, DATA_F, ei_td, ei_td + E, ea_td, E,
           P[8], P[9], P[10], P[11], P[12], P[13], P[14], P[15], true, ln_dg, ln_db, d1, B, stream);
  // layer 2 (concat=False, mean over heads + leaky 0.01)
  // td2: src=t1 (tasks), dst=d1 (data)
  run_conv(t1, NT, 64, d1, ND, 64, ei_td, ei_td + E, ea_td, E,
           P[16], P[17], P[18], P[19], P[20], P[21], P[22], P[23], false, nullptr, nullptr, d2, B,
           stream);
  // dt2: src=d1 (data), dst=t1 (tasks)
  run_conv(d1, ND, 64, t1, NT, 64, ei_dt, ei_dt + E, ea_dt, E,
           P[24], P[25], P[26], P[27], P[28], P[29], P[30], P[31], false, nullptr, nullptr, t2, B,
           stream);
}